// GCNBackbone_48189533061602
// MI455X (gfx1250) — compile-verified
//
#include <hip/hip_runtime.h>

typedef __attribute__((ext_vector_type(2))) float v2f;
typedef __attribute__((ext_vector_type(8))) float v8f;

#define D 128          // feature dim (din = dmid = dout = 128)
#define THREADS 256

// ---------------------------------------------------------------------------
// utility elementwise kernels
// ---------------------------------------------------------------------------
__global__ void k_fill(float* __restrict__ p, float val, int n) {
  int i = blockIdx.x * blockDim.x + threadIdx.x;
  if (i < n) p[i] = val;
}

// deg[col[e]] += 1  (deg pre-seeded with 1.0 for the self loop)
__global__ void k_deg_accum(const int* __restrict__ ei, float* __restrict__ deg, int E) {
  int e = blockIdx.x * blockDim.x + threadIdx.x;
  if (e < E) unsafeAtomicAdd(&deg[ei[E + e]], 1.0f);
}

// d[i] = rsqrt(d[i])   (deg >= 1 always, so no zero guard needed)
__global__ void k_rsqrt(float* __restrict__ d, int n) {
  int i = blockIdx.x * blockDim.x + threadIdx.x;
  if (i < n) d[i] = rsqrtf(d[i]);
}

// self-loop contribution: agg[i][j] = dinv[i]^2 * h[i][j]
__global__ void k_selfinit(const float* __restrict__ h, const float* __restrict__ dinv,
                           float* __restrict__ agg, int total) {
  int i = blockIdx.x * blockDim.x + threadIdx.x;
  if (i < total) {
    float dv = dinv[i >> 7];
    agg[i] = h[i] * dv * dv;
  }
}

// layer-1 epilogue: buf = relu(buf + b1) + prev_h   (in place, feeds GEMM2)
__global__ void k_post1(float* __restrict__ buf, const float* __restrict__ b,
                        const float* __restrict__ prev, int total) {
  int i = blockIdx.x * blockDim.x + threadIdx.x;
  if (i < total) {
    float v = buf[i] + b[i & (D - 1)];
    v = v > 0.0f ? v : 0.0f;
    buf[i] = v + prev[i];
  }
}

// layer-2 epilogue: out = relu(out + b2)   (in place)
__global__ void k_post2(float* __restrict__ out, const float* __restrict__ b, int total) {
  int i = blockIdx.x * blockDim.x + threadIdx.x;
  if (i < total) {
    float v = out[i] + b[i & (D - 1)];
    out[i] = v > 0.0f ? v : 0.0f;
  }
}

// ---------------------------------------------------------------------------
// edge scatter: one wave32 per edge, each lane moves 4 contiguous floats.
// agg[col] += h[row] * dinv[row]*dinv[col]   (native f32 atomics; L2-resident)
// ---------------------------------------------------------------------------
__global__ void k_scatter(const float* __restrict__ h, const int* __restrict__ ei,
                          const float* __restrict__ dinv, float* __restrict__ agg, int E) {
  int e = blockIdx.x * (blockDim.x >> 5) + (threadIdx.x >> 5);
  if (e >= E) return;
  int lane = threadIdx.x & 31;
  int row = ei[e];        // source
  int col = ei[E + e];    // destination
  float nrm = dinv[row] * dinv[col];
  const float4* src = (const float4*)(h + (size_t)row * D);
  float4 v = src[lane];                       // global_load_b128
  float* dst = agg + (size_t)col * D + lane * 4;
  unsafeAtomicAdd(dst + 0, v.x * nrm);
  unsafeAtomicAdd(dst + 1, v.y * nrm);
  unsafeAtomicAdd(dst + 2, v.z * nrm);
  unsafeAtomicAdd(dst + 3, v.w * nrm);
}

// ---------------------------------------------------------------------------
// fp32 WMMA GEMM:  C[M x 128] = A[M x 128] * W[128 x 128]
// block = 8 waves; wave w owns the 16x16 tile at columns [16w, 16w+16);
// blockIdx.x selects the 16-row stripe. K loop = 32 x V_WMMA_F32_16X16X4_F32.
// ---------------------------------------------------------------------------
__global__ void k_gemm(const float* __restrict__ A, const float* __restrict__ W,
                       float* __restrict__ C, int M) {
  int wave  = threadIdx.x >> 5;
  int lane  = threadIdx.x & 31;
  int mbase = blockIdx.x * 16;
  if (mbase >= M) return;
  int nbase = wave * 16;
  int r  = lane & 15;   // M row (A) / N col (B, C)
  int kh = lane >> 4;   // which K pair this half-wave holds

  const float* arow = A + (size_t)(mbase + r) * D;   // A fragment row
  v8f acc = {};

#pragma unroll
  for (int k0 = 0; k0 < D; k0 += 4) {
    int k = k0 + 2 * kh;
    // A 16x4: VGPR0/1 hold K = {2kh, 2kh+1} for row r  -> one aligned 8B load
    v2f a = *(const v2f*)(arow + k);
    // B 4x16: same K pair, column nbase + r (stride-D apart; L2-resident W)
    v2f b;
    b.x = W[(size_t)(k + 0) * D + nbase + r];
    b.y = W[(size_t)(k + 1) * D + nbase + r];
    acc = __builtin_amdgcn_wmma_f32_16x16x4_f32(
        /*neg_a=*/false, a, /*neg_b=*/false, b,
        /*c_mod=*/(short)0, acc, /*reuse_a=*/false, /*reuse_b=*/false);
  }

  // C/D layout: VGPR v -> M = v + 8*kh, N = r
  float* crow = C + (size_t)(mbase + 8 * kh) * D + nbase + r;
#pragma unroll
  for (int v = 0; v < 8; ++v)
    crow[(size_t)v * D] = acc[v];
}

// ---------------------------------------------------------------------------
// launch
// ---------------------------------------------------------------------------
extern "C" void kernel_launch(void* const* d_in, const int* in_sizes, int n_in,
                              void* d_out, int out_size, void* d_ws, size_t ws_size,
                              hipStream_t stream) {
  const float* x    = (const float*)d_in[0];
  const int*   ei   = (const int*)  d_in[1];   // [2][E] row-major (src, dst)
  const float* prev = (const float*)d_in[2];
  const float* W1   = (const float*)d_in[3];
  const float* b1   = (const float*)d_in[4];
  const float* W2   = (const float*)d_in[5];
  const float* b2   = (const float*)d_in[6];
  float* out = (float*)d_out;

  const int N = in_sizes[0] / D;   // 50000 nodes
  const int E = in_sizes[1] / 2;   // 600000 edges
  const int total = N * D;

  char* ws = (char*)d_ws;
  float* hbuf = (float*)ws;                               // N*D  (dense output)
  float* abuf = (float*)(ws + (size_t)total * 4);         // N*D  (aggregator)
  float* dinv = (float*)(ws + (size_t)total * 8);         // N    (deg -> dinv)

  const int T = THREADS;
  int gN  = (N + T - 1) / T;
  int gE  = (E + T - 1) / T;
  int gT  = (total + T - 1) / T;
  int gEw = (E + (T / 32) - 1) / (T / 32);   // one wave per edge
  int mtiles = (N + 15) / 16;                // 3125

  // ---- normalization: deg = 1 + rowcount(col); dinv = rsqrt(deg) ----
  k_fill     <<<gN,  T, 0, stream>>>(dinv, 1.0f, N);
  k_deg_accum<<<gE,  T, 0, stream>>>(ei, dinv, E);
  k_rsqrt    <<<gN,  T, 0, stream>>>(dinv, N);

  // ---- layer 1: f = relu(Ahat @ (x W1) + b1); buf <- f + prev_h ----
  k_gemm     <<<mtiles, 256, 0, stream>>>(x, W1, hbuf, N);
  k_selfinit <<<gT,  T, 0, stream>>>(hbuf, dinv, abuf, total);
  k_scatter  <<<gEw, T, 0, stream>>>(hbuf, ei, dinv, abuf, E);
  k_post1    <<<gT,  T, 0, stream>>>(abuf, b1, prev, total);

  // ---- layer 2: out = relu(Ahat @ ((f+prev) W2) + b2) ----
  k_gemm     <<<mtiles, 256, 0, stream>>>(abuf, W2, hbuf, N);
  k_selfinit <<<gT,  T, 0, stream>>>(hbuf, dinv, out, total);
  k_scatter  <<<gEw, T, 0, stream>>>(hbuf, ei, dinv, out, E);
  k_post2    <<<gT,  T, 0, stream>>>(out, b2, total);
}